// UniformLoss_46557445488921
// MI455X (gfx1250) — compile-verified
//
#include <hip/hip_runtime.h>
#include <math.h>

constexpr int kB    = 4;
constexpr int kN    = 8192;
constexpr int kS    = 409;        // int(8192 * 0.05)
constexpr int kG    = kB * kS;    // 1636 groups
constexpr int kTMax = 13;         // max tiles of 16 per group
constexpr int kKPad = kTMax * 16; // 208

typedef __attribute__((ext_vector_type(2))) float v2f;
typedef __attribute__((ext_vector_type(8))) float v8f;

// ---------------------------------------------------------------------------
// Kernel 1: farthest point sampling. One block per batch, 1024 threads,
// 8 points per thread held in registers together with running min-d2.
// Tie-break = lowest index (matches jnp.argmax).
// ---------------------------------------------------------------------------
#define FPS_THREADS 1024
#define FPS_PPT 8

__global__ void ul_fps_kernel(const float* __restrict__ pcs,
                              float* __restrict__ seeds) {
  const int b   = blockIdx.x;
  const int tid = threadIdx.x;
  const int lane = tid & 31;
  const int wid  = tid >> 5;
  const float* __restrict__ P = pcs + (size_t)b * kN * 3;

  float px[FPS_PPT], py[FPS_PPT], pz[FPS_PPT], md[FPS_PPT];
#pragma unroll
  for (int k = 0; k < FPS_PPT; ++k) {
    const int j = k * FPS_THREADS + tid;
    px[k] = P[j * 3 + 0];
    py[k] = P[j * 3 + 1];
    pz[k] = P[j * 3 + 2];
    md[k] = 1.0e10f;
  }

  __shared__ float swv[32];
  __shared__ int   swi[32];
  __shared__ float cur[3];
  __shared__ int   curi;

  if (tid == 0) { cur[0] = P[0]; cur[1] = P[1]; cur[2] = P[2]; }
  __syncthreads();

  for (int t = 0; t < kS; ++t) {
    const float cx = cur[0], cy = cur[1], cz = cur[2];
    if (tid == 0) {
      seeds[((size_t)b * kS + t) * 3 + 0] = cx;
      seeds[((size_t)b * kS + t) * 3 + 1] = cy;
      seeds[((size_t)b * kS + t) * 3 + 2] = cz;
    }
    float bv = -1.0f;
    int   bi = 0x7fffffff;
#pragma unroll
    for (int k = 0; k < FPS_PPT; ++k) {
      const float dx = px[k] - cx, dy = py[k] - cy, dz = pz[k] - cz;
      const float d = dx * dx + dy * dy + dz * dz;
      md[k] = fminf(md[k], d);
      if (md[k] > bv) { bv = md[k]; bi = k * FPS_THREADS + tid; }
    }
    // wave reduce (max value, min index on ties)
#pragma unroll
    for (int off = 16; off; off >>= 1) {
      const float ov = __shfl_xor(bv, off, 32);
      const int   oi = __shfl_xor(bi, off, 32);
      if (ov > bv || (ov == bv && oi < bi)) { bv = ov; bi = oi; }
    }
    if (lane == 0) { swv[wid] = bv; swi[wid] = bi; }
    __syncthreads();
    if (wid == 0) {
      bv = swv[lane];
      bi = swi[lane];
#pragma unroll
      for (int off = 16; off; off >>= 1) {
        const float ov = __shfl_xor(bv, off, 32);
        const int   oi = __shfl_xor(bi, off, 32);
        if (ov > bv || (ov == bv && oi < bi)) { bv = ov; bi = oi; }
      }
      if (lane == 0) curi = bi;
    }
    __syncthreads();
    const int ci = curi;
    if (tid == (ci & (FPS_THREADS - 1))) {
      const int k = ci >> 10;
      cur[0] = px[k]; cur[1] = py[k]; cur[2] = pz[k];
    }
    __syncthreads();
  }
}

// ---------------------------------------------------------------------------
// Kernel 2 (per p): ball query with ordered compaction. One wave per seed.
// Writes up to K points (index order), sentinel-pads to Kpad, stores gnum and
// imbalance/gnum term.
// ---------------------------------------------------------------------------
__global__ void ul_build_groups_kernel(const float* __restrict__ pcs,
                                       const float* __restrict__ seeds,
                                       float* __restrict__ pts,      // kG * kKPad * 3
                                       int*   __restrict__ gnum,     // kG
                                       float* __restrict__ imbTerm,  // kG (this p)
                                       float rr, float expnum, int K, int Kpad) {
  const int gw   = (blockIdx.x * blockDim.x + threadIdx.x) >> 5;
  const int lane = threadIdx.x & 31;
  if (gw >= kG) return;

  const float sx = seeds[gw * 3 + 0];
  const float sy = seeds[gw * 3 + 1];
  const float sz = seeds[gw * 3 + 2];
  const int b = gw / kS;
  const float* __restrict__ P = pcs + (size_t)b * kN * 3;
  float* __restrict__ gp = pts + (size_t)gw * kKPad * 3;

  int cnt = 0;
  for (int base = 0; base < kN; base += 32) {
    const int j = base + lane;
    const float x = P[j * 3 + 0];
    const float y = P[j * 3 + 1];
    const float z = P[j * 3 + 2];
    const float dx = x - sx, dy = y - sy, dz = z - sz;
    const float d2 = dx * dx + dy * dy + dz * dz;
    const bool valid = d2 < rr;
    const unsigned m = (unsigned)__builtin_amdgcn_ballot_w32(valid);
    if (valid) {
      const int pos = cnt + __popc(m & ((1u << lane) - 1u));
      if (pos < K) {
        gp[pos * 3 + 0] = x;
        gp[pos * 3 + 1] = y;
        gp[pos * 3 + 2] = z;
      }
    }
    cnt += __popc(m);
  }
  const int gn = cnt < K ? cnt : K;
  // sentinel padding: far-away distinct points so they never win a min
  for (int slot = gn + lane; slot < Kpad; slot += 32) {
    const float sv = 1.0e5f * (float)(slot + 1);
    gp[slot * 3 + 0] = sv;
    gp[slot * 3 + 1] = sv;
    gp[slot * 3 + 2] = sv;
  }
  if (lane == 0) {
    gnum[gw] = gn;
    const float g   = (float)gn;
    const float dlt = g - expnum;
    imbTerm[gw] = (dlt * dlt / expnum) / g;
  }
}

// ---------------------------------------------------------------------------
// Kernel 3 (per p): 2nd-NN squared distance per group member via fp32 WMMA.
// 16x16x4 tiles with the row norm folded into the matmul:
//   A row m = (-2ax, -2ay, -2az, |a|^2),  B col n = (bx, by, bz, 1)
//   => D[m][n] = |a_m|^2 - 2 a_m.b_n   (add |b_n|^2 once per column at end)
// One wave per (group, column-tile J). No cross-lane traffic in the loop.
// ---------------------------------------------------------------------------
__global__ void ul_knn_clutter_kernel(const float* __restrict__ pts,
                                      const int*   __restrict__ gnum,
                                      float* __restrict__ partial, // kG * kTMax (this p)
                                      float expect, int Tp) {
  const int gw   = (blockIdx.x * blockDim.x + threadIdx.x) >> 5;
  const int lane = threadIdx.x & 31;
  const int total = kG * Tp;
  if (gw >= total) return;

  const int g = gw / Tp;
  const int J = gw % Tp;
  const int gn = gnum[g];
  const int Jbase = J * 16;

  float cl = 0.0f;
  if (Jbase < gn) {
    const float* __restrict__ gp = pts + (size_t)g * kKPad * 3;
    const int col = lane & 15;
    const bool lowHalf = (lane < 16);
    const int n = Jbase + col;
    const float bx = gp[n * 3 + 0];
    const float by = gp[n * 3 + 1];
    const float bz = gp[n * 3 + 2];
    const float normB = bx * bx + by * by + bz * bz;
    v2f Bop;                               // B 4x16: K0/K2 in v0, K1/K3 in v1
    Bop.x = lowHalf ? bx : bz;
    Bop.y = lowHalf ? by : 1.0f;

    const int half8 = lowHalf ? 0 : 8;
    float runmin = 3.0e38f;
    const int Tg = (gn + 15) >> 4;
    for (int I = 0; I < Tg; ++I) {
      const int m = I * 16 + col;
      const float ax = gp[m * 3 + 0];
      const float ay = gp[m * 3 + 1];
      const float az = gp[m * 3 + 2];
      const float normA = ax * ax + ay * ay + az * az;
      v2f Aop;                             // A 16x4: lanes<16 -> K0/K1, >=16 -> K2/K3
      Aop.x = lowHalf ? (-2.0f * ax) : (-2.0f * az);
      Aop.y = lowHalf ? (-2.0f * ay) : normA;
      v8f c = {};
      c = __builtin_amdgcn_wmma_f32_16x16x4_f32(
          /*neg_a=*/false, Aop, /*neg_b=*/false, Bop,
          /*c_mod=*/(short)0, c, /*reuse_a=*/false, /*reuse_b=*/false);
      if (I == J) {                        // wave-uniform: mask the self term
#pragma unroll
        for (int v = 0; v < 8; ++v) {
          if ((v + half8) != col) runmin = fminf(runmin, c[v]);
        }
      } else {
#pragma unroll
        for (int v = 0; v < 8; ++v) runmin = fminf(runmin, c[v]);
      }
    }
    runmin = fminf(runmin, __shfl_xor(runmin, 16, 32)); // combine M halves
    if (lowHalf && n < gn) {
      const float nn2 = runmin + normB;
      // sentinels / no-neighbor (gn==1) -> huge value -> treated as nn_dis = 0
      const float nn_dis = (nn2 > 0.0f && nn2 < 1.0e8f) ? sqrtf(nn2) : 0.0f;
      const float diff = nn_dis - expect;
      cl = diff * diff / (expect + 1e-12f);
    }
  }
#pragma unroll
  for (int off = 16; off; off >>= 1) cl += __shfl_xor(cl, off, 32);
  if (lane == 0) partial[g * kTMax + J] = cl;
}

// ---------------------------------------------------------------------------
// Kernel 4: deterministic final reduction.
// loss[b] = (1/5) * sum_p S_total(p) * mean_s( imbalance/gnum )
// ---------------------------------------------------------------------------
__global__ void ul_finalize_kernel(const float* __restrict__ imbTerm, // 5*kG
                                   const float* __restrict__ partial, // 5*kG*kTMax
                                   float* __restrict__ out) {
  __shared__ float red[256];
  __shared__ float acc[kB];
  const int tid = threadIdx.x;
  if (tid < kB) acc[tid] = 0.0f;
  __syncthreads();

  constexpr int Ts[5] = {5, 7, 9, 11, 13};
  for (int p = 0; p < 5; ++p) {
    // global clutter sum for this p
    float s = 0.0f;
    const float* pp = partial + (size_t)p * kG * kTMax;
    for (int i = tid; i < kG * kTMax; i += 256) {
      if ((i % kTMax) < Ts[p]) s += pp[i];
    }
    red[tid] = s;
    __syncthreads();
    for (int st = 128; st; st >>= 1) {
      if (tid < st) red[tid] += red[tid + st];
      __syncthreads();
    }
    const float Stot = red[0];
    __syncthreads();

    for (int b = 0; b < kB; ++b) {
      float s2 = 0.0f;
      const float* it = imbTerm + (size_t)p * kG + b * kS;
      for (int i = tid; i < kS; i += 256) s2 += it[i];
      red[tid] = s2;
      __syncthreads();
      for (int st = 128; st; st >>= 1) {
        if (tid < st) red[tid] += red[tid + st];
        __syncthreads();
      }
      if (tid == 0) acc[b] += Stot * (red[0] / (float)kS);
      __syncthreads();
    }
  }
  if (tid < kB) out[tid] = acc[tid] / 5.0f;
}

// ---------------------------------------------------------------------------
extern "C" void kernel_launch(void* const* d_in, const int* in_sizes, int n_in,
                              void* d_out, int out_size, void* d_ws, size_t ws_size,
                              hipStream_t stream) {
  (void)in_sizes; (void)n_in; (void)out_size; (void)ws_size;
  const float* pcs = (const float*)d_in[0];
  float* out = (float*)d_out;

  char* ws = (char*)d_ws;
  size_t off = 0;
  auto walloc = [&](size_t bytes) -> void* {
    void* p = ws + off;
    off += (bytes + 255) & ~(size_t)255;
    return p;
  };
  float* seeds    = (float*)walloc((size_t)kB * kS * 3 * sizeof(float));
  int*   gnum     = (int*)  walloc((size_t)kG * sizeof(int));
  float* imbTerm  = (float*)walloc((size_t)5 * kG * sizeof(float));
  float* partial  = (float*)walloc((size_t)5 * kG * kTMax * sizeof(float));
  float* pts      = (float*)walloc((size_t)kG * kKPad * 3 * sizeof(float));

  ul_fps_kernel<<<kB, FPS_THREADS, 0, stream>>>(pcs, seeds);

  const double Ps[5] = {0.004, 0.006, 0.008, 0.01, 0.012};
  const int    Ks[5] = {65, 98, 131, 163, 196};  // int(2 * 8192 * p)
  const int    Ts[5] = {5, 7, 9, 11, 13};        // ceil(K/16)

  for (int p = 0; p < 5; ++p) {
    const double r = sqrt(Ps[p]);
    const float rr = (float)(r * r);
    const float expnum = (float)(8192.0 * Ps[p]);
    const float expect = sqrtf((float)(3.14159265358979323846 / (8192.0 * Ps[p]) * Ps[p]));
    const int K = Ks[p], T = Ts[p], Kpad = T * 16;

    const int buildBlocks = (kG + 7) / 8;  // 8 waves / 256-thread block
    ul_build_groups_kernel<<<buildBlocks, 256, 0, stream>>>(
        pcs, seeds, pts, gnum, imbTerm + (size_t)p * kG, rr, expnum, K, Kpad);

    const int knnWaves  = kG * T;
    const int knnBlocks = (knnWaves + 7) / 8;
    ul_knn_clutter_kernel<<<knnBlocks, 256, 0, stream>>>(
        pts, gnum, partial + (size_t)p * kG * kTMax, expect, T);
  }

  ul_finalize_kernel<<<1, 256, 0, stream>>>(imbTerm, partial, out);
}